// GCN_63350767616684
// MI455X (gfx1250) — compile-verified
//
#include <hip/hip_runtime.h>
#include <math.h>

typedef __attribute__((ext_vector_type(2))) float v2f;
typedef __attribute__((ext_vector_type(8))) float v8f;

// ---------------------------------------------------------------- utilities
__global__ void zero_f32(float* __restrict__ p, long n) {
    long i = (long)blockIdx.x * blockDim.x + threadIdx.x;
    if (i < n) p[i] = 0.0f;
}

__global__ void degree_kernel(const int* __restrict__ src, const int* __restrict__ dst,
                              float* __restrict__ deg_out, float* __restrict__ deg_in, int E) {
    int e = blockIdx.x * blockDim.x + threadIdx.x;
    if (e < E) {
        atomicAdd(&deg_out[src[e]], 1.0f);
        atomicAdd(&deg_in[dst[e]], 1.0f);
    }
}

// deg -> deg^-0.5 (0 stays 0); applied to both ns and nd in one pass
__global__ void norm_kernel(float* __restrict__ p, int n) {
    int i = blockIdx.x * blockDim.x + threadIdx.x;
    if (i < n) {
        float d = p[i];
        p[i] = (d > 0.0f) ? rsqrtf(d) : 0.0f;
    }
}

// ------------------------------------------------- GEMM: out = (X @ W) * ns
// One wave per 16x16 output tile. fp32 WMMA, K stepped by 4, unrolled 4x.
// A layout (16x4 f32): lanes 0-15 (row M=lane) hold K=0/K=1, lanes 16-31 hold K=2/K=3.
// B layout (4x16 f32): symmetric, lane owns column N=lane&15.
// D layout: VGPR i -> row (i + 8*(lane>>4)), col = lane&15 within tile.
// GUARD=false: all tile columns valid (M multiple of 16) -> zero predication.
// GUARD=true : clamp column to M-1 (always-in-bounds load) and multiply by 0/1
//              mask, so EXEC stays all-ones through the whole loop (no saveexec).
template <bool GUARD>
__global__ void gemm_ns_wmma(const float* __restrict__ X, const float* __restrict__ W,
                             const float* __restrict__ ns, float* __restrict__ out,
                             int N, int K, int M, int col_tiles) {
    const int wave          = threadIdx.x >> 5;
    const int lane          = threadIdx.x & 31;
    const int waves_per_blk = blockDim.x >> 5;
    const int row_tiles     = N >> 4;
    const int n_tiles       = row_tiles * col_tiles;

    int tile = blockIdx.x * waves_per_blk + wave;   // uniform within wave
    if (tile >= n_tiles) return;                    // whole-wave exit, EXEC stays full

    const int rt   = tile / col_tiles;
    const int ct   = tile - rt * col_tiles;
    const int half = lane >> 4;        // 0: K pair {0,1}, 1: K pair {2,3}
    const int lid  = lane & 15;

    const int row = (rt << 4) + lid;   // A row owned by this lane (rows always in range)
    const int col = (ct << 4) + lid;   // B/D column owned by this lane

    int   colc  = col;
    float bmask = 1.0f;
    if (GUARD && col >= M) { colc = M - 1; bmask = 0.0f; }   // clamped, in-bounds

    const float* xrow = X + (long)row * K;
    const float* wcol = W + colc;      // column base; element k at wcol[k*M]
    v8f acc = {};

#pragma unroll 4
    for (int kk = 0; kk < K; kk += 4) {
        const int ka = kk + 2 * half;
        v2f a, b;
        a.x = xrow[ka];
        a.y = xrow[ka + 1];
        b.x = wcol[(long)ka * M];
        b.y = wcol[(long)(ka + 1) * M];
        if (GUARD) { b.x *= bmask; b.y *= bmask; }
        // 8 args: (neg_a, A, neg_b, B, c_mod, C, reuse_a, reuse_b)
        acc = __builtin_amdgcn_wmma_f32_16x16x4_f32(
            false, a, false, b, (short)0, acc, false, false);
    }

    if (!GUARD || col < M) {
#pragma unroll
        for (int i = 0; i < 8; ++i) {
            const int orow = (rt << 4) + i + 8 * half;
            out[(long)orow * M + col] = acc[i] * ns[orow];
        }
    }
}

// ------------------------------------------- SpMM: agg[dst] += tmp[src] (edge||)
// Each thread handles one edge x 4 features (float4 gather, 4 scalar float atomics).
// All gather/scatter traffic is L2-resident (working set << 192 MB).
__global__ void spmm_scatter(const float* __restrict__ tmp, const int* __restrict__ src,
                             const int* __restrict__ dst, float* __restrict__ agg,
                             int E, int M) {
    const int  chunks = M >> 2;                  // M is a multiple of 4 (64, 40)
    const long t      = (long)blockIdx.x * blockDim.x + threadIdx.x;
    const long total  = (long)E * chunks;
    if (t >= total) return;
    const int e = (int)(t / chunks);
    const int f = (int)(t - (long)e * chunks) << 2;

    const float4 v = *(const float4*)(tmp + (long)src[e] * M + f);
    float* d = agg + (long)dst[e] * M + f;
    atomicAdd(d + 0, v.x);
    atomicAdd(d + 1, v.y);
    atomicAdd(d + 2, v.z);
    atomicAdd(d + 3, v.w);
}

// --------------------- out = log_softmax(relu?(agg*nd + b)) ; one wave per row
template <bool RELU>
__global__ void post_logsoftmax(const float* __restrict__ agg, const float* __restrict__ nd,
                                const float* __restrict__ bias, float* __restrict__ out,
                                int N, int M) {
    const int wave = threadIdx.x >> 5;
    const int lane = threadIdx.x & 31;
    const int row  = blockIdx.x * (blockDim.x >> 5) + wave;
    if (row >= N) return;

    const float  s    = nd[row];
    const float* arow = agg + (long)row * M;

    float x0 = -INFINITY, x1 = -INFINITY;
    const int l1 = lane, l2 = lane + 32;     // M <= 64
    if (l1 < M) { x0 = arow[l1] * s + bias[l1]; if (RELU) x0 = fmaxf(x0, 0.0f); }
    if (l2 < M) { x1 = arow[l2] * s + bias[l2]; if (RELU) x1 = fmaxf(x1, 0.0f); }

    float m = fmaxf(x0, x1);
    for (int o = 16; o > 0; o >>= 1) m = fmaxf(m, __shfl_xor(m, o, 32));

    float sum = 0.0f;
    if (l1 < M) sum += expf(x0 - m);
    if (l2 < M) sum += expf(x1 - m);
    for (int o = 16; o > 0; o >>= 1) sum += __shfl_xor(sum, o, 32);

    const float lse = m + logf(sum);
    float* orow = out + (long)row * M;
    if (l1 < M) orow[l1] = x0 - lse;
    if (l2 < M) orow[l2] = x1 - lse;
}

// --------------------------------------------------------------------------
extern "C" void kernel_launch(void* const* d_in, const int* in_sizes, int n_in,
                              void* d_out, int out_size, void* d_ws, size_t ws_size,
                              hipStream_t stream) {
    const float* feats = (const float*)d_in[0];
    const int*   src   = (const int*)d_in[1];   // JAX default x64-off -> int32 on device
    const int*   dst   = (const int*)d_in[2];
    const float* W0    = (const float*)d_in[3];
    const float* b0    = (const float*)d_in[4];
    const float* W1    = (const float*)d_in[5];
    const float* b1    = (const float*)d_in[6];
    const float* W2    = (const float*)d_in[7];
    const float* b2    = (const float*)d_in[8];

    const int N = in_sizes[0] / 128;   // 100000 (divisible by 16)
    const int E = in_sizes[1];         // 1600000

    float* ws   = (float*)d_ws;
    float* ns   = ws;                  // [N]
    float* nd   = ws + N;              // [N]
    float* bufA = ws + 2L * N;         // [N*64]
    float* bufB = bufA + (long)N * 64; // [N*64]

    // ---- norms: zero, degree atomics, rsqrt transform
    {
        long n = 2L * N;
        zero_f32<<<(unsigned)((n + 255) / 256), 256, 0, stream>>>(ns, n);
    }
    degree_kernel<<<(E + 255) / 256, 256, 0, stream>>>(src, dst, ns, nd, E);
    norm_kernel<<<(2 * N + 255) / 256, 256, 0, stream>>>(ns, 2 * N);

    const int row_tiles = N / 16;
    auto gemm_full = [&](const float* X, const float* W, float* o, int K, int M) {
        const int ct     = M / 16;
        const int tiles  = row_tiles * ct;
        const int blocks = (tiles + 7) / 8;        // 8 waves (256 threads) per block
        gemm_ns_wmma<false><<<blocks, 256, 0, stream>>>(X, W, ns, o, N, K, M, ct);
    };
    auto gemm_guard = [&](const float* X, const float* W, float* o, int K, int M) {
        const int ct     = (M + 15) / 16;
        const int tiles  = row_tiles * ct;
        const int blocks = (tiles + 7) / 8;
        gemm_ns_wmma<true><<<blocks, 256, 0, stream>>>(X, W, ns, o, N, K, M, ct);
    };
    auto zero = [&](float* p, long n) {
        zero_f32<<<(unsigned)((n + 255) / 256), 256, 0, stream>>>(p, n);
    };
    auto spmm = [&](const float* tmp, float* agg, int M) {
        const long total  = (long)E * (M / 4);
        const long blocks = (total + 255) / 256;
        spmm_scatter<<<(unsigned)blocks, 256, 0, stream>>>(tmp, src, dst, agg, E, M);
    };

    // ---- layer 0: 128 -> 64, relu, log_softmax
    gemm_full(feats, W0, bufA, 128, 64);
    zero(bufB, (long)N * 64);
    spmm(bufA, bufB, 64);
    post_logsoftmax<true><<<(N + 7) / 8, 256, 0, stream>>>(bufB, nd, b0, bufA, N, 64);

    // ---- layer 1: 64 -> 64, relu, log_softmax
    gemm_full(bufA, W1, bufB, 64, 64);
    zero(bufA, (long)N * 64);
    spmm(bufB, bufA, 64);
    post_logsoftmax<true><<<(N + 7) / 8, 256, 0, stream>>>(bufA, nd, b1, bufB, N, 64);

    // ---- layer 2: 64 -> 40, log_softmax -> d_out
    gemm_guard(bufB, W2, bufA, 64, 40);
    zero(bufB, (long)N * 40);
    spmm(bufA, bufB, 40);
    post_logsoftmax<false><<<(N + 7) / 8, 256, 0, stream>>>(bufB, nd, b2, (float*)d_out, N, 40);
}